// SSIMLoss3D_33414845563615
// MI455X (gfx1250) — compile-verified
//
#include <hip/hip_runtime.h>

// ---------------------------------------------------------------------------
// SSIM-3D loss, fully fused for gfx1250 (MI455X).
//   - separable 7-tap Gaussian conv as banded-matrix WMMA (f32 16x16x4)
//   - 5 fields (p, t, p^2, t^2, p*t) fused, all intermediates in LDS
//   - hot LDS buffers in low 64KB so DS immediate offsets fold
//   - scalar (readfirstlane) tile loops: no exec juggling, EXEC==~0 at WMMA
//   - single HBM pass over inputs, deterministic 2-kernel reduction
// ---------------------------------------------------------------------------

typedef float v2f __attribute__((ext_vector_type(2)));
typedef float v8f __attribute__((ext_vector_type(8)));

#define DIM      160
#define TILE     16
#define HALO     22                      // TILE + 6
#define NHALO    (HALO*HALO*HALO)        // 10648
#define BUFX_L   496                     // padded to 31 full 16-line tiles
#define BUFX_N   (BUFX_L*TILE)           // 7936
#define BUFY_N   (HALO*TILE*TILE)        // 5632
#define NTOTAL   (4*DIM*DIM*DIM)         // 16,384,000
#define NBLOCKS  (10*10*40)              // 4000

// normalized Gaussian(sigma=1.5) 7-tap weights
__device__ const float GW7[7] = {
    0.03663284531113f, 0.11128075506474f, 0.21674532140370f,
    0.27068215774086f, 0.21674532140370f, 0.11128075506474f,
    0.03663284531113f };

// LDS float offsets: hot buffers first (all DS offsets < 64KB fold to imms)
#define OFF_BX   0                        // 7936  (bytes     0..31743)
#define OFF_BY   (OFF_BX + BUFX_N)        // 7936  (bytes 31744..54271)
#define OFF_RED  (OFF_BY + BUFY_N)        // 13568
#define OFF_GW   (OFF_RED + 256)          // 13824
#define OFF_P    (OFF_GW + 8)             // 13832
#define OFF_T    (OFF_P + NHALO)          // 24480
#define LDS_F    (OFF_T + NHALO + 272)    // 35400 (272-float overread pad)
#define LDS_BYTES (LDS_F * 4)             // 141600 B (< 320 KB WGP LDS)

__device__ __forceinline__ float field_op(int f, float p, float t) {
    switch (f) {
        case 0:  return p;
        case 1:  return t;
        case 2:  return p * p;
        case 3:  return t * t;
        default: return p * t;
    }
}

__launch_bounds__(256)
__global__ void ssim3d_main(const float* __restrict__ pred,
                            const float* __restrict__ tgt,
                            float* __restrict__ partial) {
    extern __shared__ float smem[];
    float* bufX = smem + OFF_BX;          // [line(z,y) 0..495][x_out 16]
    float* bufY = smem + OFF_BY;          // transposed: [z 22][x 16][y_out 16]
    float* red  = smem + OFF_RED;
    float* gw   = smem + OFF_GW;
    float* sP   = smem + OFF_P;
    float* sT   = smem + OFF_T;

    const int tid  = threadIdx.x;
    const int lane = tid & 31;
    const int wvu  = __builtin_amdgcn_readfirstlane(tid >> 5);  // scalar wave id
    const int hi   = (lane >> 4) & 1;     // lane half (range-known for folding)
    const int mn   = lane & 15;           // M (for A) / N (for B,C) index

    const int batch = blockIdx.z / 10;
    const int tz    = blockIdx.z % 10;
    const int x0 = blockIdx.x * TILE - 3;
    const int y0 = blockIdx.y * TILE - 3;
    const int z0 = tz * TILE - 3;
    const int gbase = batch * (DIM * DIM * DIM);

    if (tid < 7) gw[tid] = GW7[tid];

    // ---- load 22^3 halos of p,t into LDS (fast path for interior blocks) ----
    const bool interior = (x0 >= 0) && (y0 >= 0) && (z0 >= 0) &&
                          (x0 + HALO <= DIM) && (y0 + HALO <= DIM) &&
                          (z0 + HALO <= DIM);
    if (interior) {
        for (int idx = tid; idx < NHALO; idx += 256) {
            int i  = idx % HALO;
            int r  = idx / HALO;
            int j  = r % HALO;
            int k  = r / HALO;
            int gidx = gbase + ((z0 + k) * DIM + (y0 + j)) * DIM + (x0 + i);
            sP[idx] = pred[gidx];
            sT[idx] = tgt[gidx];
        }
    } else {
        for (int idx = tid; idx < NHALO; idx += 256) {
            int i  = idx % HALO;
            int r  = idx / HALO;
            int j  = r % HALO;
            int k  = r / HALO;
            int gx = x0 + i, gy = y0 + j, gz = z0 + k;
            bool inb = (gx >= 0) && (gx < DIM) && (gy >= 0) && (gy < DIM) &&
                       (gz >= 0) && (gz < DIM);
            int gidx = gbase + (gz * DIM + gy) * DIM + gx;
            sP[idx] = inb ? pred[gidx] : 0.0f;
            sT[idx] = inb ? tgt[gidx]  : 0.0f;
        }
    }
    __syncthreads();

    // ---- precompute banded Gaussian A-matrix chunks (W[m][k] = g[k-m]) ----
    // A 16x4 layout: lanes 0-15 M=0..15 / VGPR0:{K0,K2}, VGPR1:{K1,K3}
    v2f Amat[6];
#pragma unroll
    for (int kc = 0; kc < 6; ++kc) {
        int k0 = kc * 4 + 2 * hi;
        int d0 = k0 - mn, d1 = k0 + 1 - mn;
        Amat[kc].x = (k0     < HALO && d0 >= 0 && d0 < 7) ? gw[d0] : 0.0f;
        Amat[kc].y = (k0 + 1 < HALO && d1 >= 0 && d1 < 7) ? gw[d1] : 0.0f;
    }

    v8f acc[5][2];   // persistent conv results: 5 fields x 2 C-tiles per wave

#pragma unroll
    for (int f = 0; f < 5; ++f) {
        // ---- pass X: conv along x.  out bufX[line(z,y)][x_out] ----
        // All loads unconditional: clamped K (A is zero for k>=22); line
        // overreads land in finite LDS (sT / pad); garbage C columns go to
        // bufX padding lines (484..495) which pass Y never reads.
        for (int t = wvu; t < 31; t += 8) {       // 31 tiles of 16 lines
            const int l0 = t * 16;
            const int lb = (l0 + mn) * HALO;
            v8f c = {0.f,0.f,0.f,0.f,0.f,0.f,0.f,0.f};
#pragma unroll
            for (int kc = 0; kc < 6; ++kc) {
                int kA  = kc * 4 + 2 * hi;
                int kAc = (kA > 20) ? 20 : kA;    // folds away for kc<5
                v2f P = *(const v2f*)(sP + lb + kAc);   // ds_load_b64 (merged)
                v2f T = *(const v2f*)(sT + lb + kAc);   // ds_load_b64 (merged)
                v2f b;
                b.x = field_op(f, P.x, T.x);
                b.y = field_op(f, P.y, T.y);
                c = __builtin_amdgcn_wmma_f32_16x16x4_f32(
                        false, Amat[kc], false, b, (short)0, c, false, false);
            }
            // contiguous 8-float store, 32B aligned -> 2x ds_store_b128
            *(v8f*)(bufX + (l0 + mn) * TILE + 8 * hi) = c;
        }
        __syncthreads();

        // ---- pass Y: conv along y.  bufX[z][y][x] -> bufY[z][x][y_out] ----
        for (int z = wvu; z < HALO; z += 8) {     // 22 z-slices
            const float* bx = bufX + z * (HALO * TILE) + mn;
            v8f c = {0.f,0.f,0.f,0.f,0.f,0.f,0.f,0.f};
#pragma unroll
            for (int kc = 0; kc < 6; ++kc) {
                int kA  = kc * 4 + 2 * hi;
                int kAc = (kA > 20) ? 20 : kA;
                v2f b;
                b.x = bx[kAc * TILE];             // imm offsets off one base
                b.y = bx[(kAc + 1) * TILE];
                c = __builtin_amdgcn_wmma_f32_16x16x4_f32(
                        false, Amat[kc], false, b, (short)0, c, false, false);
            }
            // transposed store: contiguous in y_out -> 2x ds_store_b128
            *(v8f*)(bufY + z * 256 + mn * TILE + 8 * hi) = c;
        }
        __syncthreads();

        // ---- pass Z: conv along z, accumulate into registers ----
#pragma unroll
        for (int s = 0; s < 2; ++s) {
            const int yq = wvu + 8 * s;           // y of this C tile (scalar)
            const float* by = bufY + mn * TILE + yq;
            v8f c = {0.f,0.f,0.f,0.f,0.f,0.f,0.f,0.f};
#pragma unroll
            for (int kc = 0; kc < 6; ++kc) {
                int kA  = kc * 4 + 2 * hi;
                int kAc = (kA > 20) ? 20 : kA;
                v2f b;
                b.x = by[kAc * 256];              // imm offsets off one base
                b.y = by[(kAc + 1) * 256];
                c = __builtin_amdgcn_wmma_f32_16x16x4_f32(
                        false, Amat[kc], false, b, (short)0, c, false, false);
            }
            acc[f][s] = c;
        }
        __syncthreads();   // protect bufX/bufY before next field
    }

    // ---- SSIM map + block partial sum (each lane owns 16 voxels) ----
    const float C1 = 1.0e-4f;   // 0.01^2
    const float C2 = 9.0e-4f;   // 0.03^2
    float lsum = 0.0f;
#pragma unroll
    for (int s = 0; s < 2; ++s) {
#pragma unroll
        for (int r = 0; r < 8; ++r) {
            float mu_p = acc[0][s][r];
            float mu_t = acc[1][s][r];
            float spp  = acc[2][s][r] - mu_p * mu_p;
            float stt  = acc[3][s][r] - mu_t * mu_t;
            float spt  = acc[4][s][r] - mu_p * mu_t;
            float num  = (2.0f * mu_p * mu_t + C1) * (2.0f * spt + C2);
            float den  = (mu_p * mu_p + mu_t * mu_t + C1) * (spp + stt + C2);
            lsum += num / den;
        }
    }
    red[tid] = lsum;
    __syncthreads();
#pragma unroll
    for (int off = 128; off > 0; off >>= 1) {
        if (tid < off) red[tid] += red[tid + off];
        __syncthreads();
    }
    if (tid == 0)
        partial[blockIdx.x + 10 * blockIdx.y + 100 * blockIdx.z] = red[0];
}

__launch_bounds__(256)
__global__ void ssim3d_reduce(const float* __restrict__ partial,
                              float* __restrict__ out) {
    __shared__ float red[256];
    const int tid = threadIdx.x;
    float s = 0.0f;
    for (int i = tid; i < NBLOCKS; i += 256) s += partial[i];
    red[tid] = s;
    __syncthreads();
#pragma unroll
    for (int off = 128; off > 0; off >>= 1) {
        if (tid < off) red[tid] += red[tid + off];
        __syncthreads();
    }
    if (tid == 0) out[0] = 1.0f - red[0] / (float)NTOTAL;
}

extern "C" void kernel_launch(void* const* d_in, const int* in_sizes, int n_in,
                              void* d_out, int out_size, void* d_ws, size_t ws_size,
                              hipStream_t stream) {
    const float* pred = (const float*)d_in[0];
    const float* tgt  = (const float*)d_in[1];
    float* out        = (float*)d_out;
    float* partial    = (float*)d_ws;      // 4000 floats of scratch

    // allow >64KB dynamic LDS (CDNA5 WGP has 320KB)
    hipFuncSetAttribute((const void*)ssim3d_main,
                        hipFuncAttributeMaxDynamicSharedMemorySize, LDS_BYTES);

    dim3 grid(10, 10, 40);   // 10x10 tiles in x,y ; 10 z-tiles x 4 batches
    dim3 block(256);
    ssim3d_main<<<grid, block, LDS_BYTES, stream>>>(pred, tgt, partial);
    ssim3d_reduce<<<1, 256, 0, stream>>>(partial, out);
}